// LRU_15032385536244
// MI455X (gfx1250) — compile-verified
//
#include <hip/hip_runtime.h>

typedef float v2f __attribute__((ext_vector_type(2)));
typedef float v8f __attribute__((ext_vector_type(8)));

#define B_SZ   262144
#define IN_DIM 128
#define HID    256
#define NTOT   512   // [0,256)=real columns, [256,512)=imag columns

// ---------------------------------------------------------------------------
// Prep: build swizzled weight Wp2 (B-operand layout for V_WMMA_F32_16X16X4_F32)
// and lam_re/lam_im tables.
//   Wp2[(c*512 + n)*2 + j] = { W[4c+2j][n], W[4c+2j+1][n] }
//   W[k][n] = exp(gamma_log[n&255]) * (n<256 ? B_real : B_img)[n&255][k]
// ---------------------------------------------------------------------------
__global__ void lru_prep(const float* __restrict__ nu_log,
                         const float* __restrict__ theta_log,
                         const float* __restrict__ B_real,
                         const float* __restrict__ B_img,
                         const float* __restrict__ gamma_log,
                         float2* __restrict__ Wp2,
                         float*  __restrict__ lam)
{
    int t = blockIdx.x * blockDim.x + threadIdx.x;
    if (t < 32 * 512 * 2) {               // 32768 float2 entries
        int j = t & 1;
        int n = (t >> 1) & 511;
        int c = t >> 10;
        int k = 4 * c + 2 * j;
        int h = n & 255;
        const float* src = (n < HID) ? B_real : B_img;
        float s = expf(gamma_log[h]);
        Wp2[t] = make_float2(s * src[h * IN_DIM + k],
                             s * src[h * IN_DIM + k + 1]);
    } else if (t < 32768 + HID) {
        int h = t - 32768;
        float mag = expf(-expf(nu_log[h]));
        float th  = expf(theta_log[h]);
        lam[h]        = mag * cosf(th);   // lam_re
        lam[HID + h]  = mag * sinf(th);   // lam_im
    }
}

// ---------------------------------------------------------------------------
// Main: 256 threads = 8 waves. Block -> 64 rows. Wave w -> 64-col N strip.
// K streamed in chunks of 4 through v_wmma_f32_16x16x4_f32.
// ---------------------------------------------------------------------------
__global__ __launch_bounds__(256)
void lru_main(const float* __restrict__ inputs,
              const float* __restrict__ h_re,
              const float* __restrict__ h_im,
              const float* __restrict__ Wp2f,   // swizzled weights (as float*)
              const float* __restrict__ lam,
              float* __restrict__ out)
{
    const int lane = threadIdx.x & 31;
    const int wave = threadIdx.x >> 5;
    const int lhi  = lane >> 4;        // 0 or 1 (half-wave)
    const int llo  = lane & 15;
    const int m0   = blockIdx.x * 64;
    const int n0   = wave * 64;

    // A operand address (elements): row = m0+ms*16+llo, col base = lhi*2
    int aBase[4];
#pragma unroll
    for (int ms = 0; ms < 4; ++ms)
        aBase[ms] = (m0 + ms * 16 + llo) * IN_DIM + lhi * 2;

    // B operand address in float2 units: (n)*2 + lhi, stepping 1024 per k-chunk
    int bBase[4];
#pragma unroll
    for (int nt = 0; nt < 4; ++nt)
        bBase[nt] = (n0 + nt * 16 + llo) * 2 + lhi;

    v8f acc[4][4];
    const v8f zero = {};
#pragma unroll
    for (int ms = 0; ms < 4; ++ms)
#pragma unroll
        for (int nt = 0; nt < 4; ++nt)
            acc[ms][nt] = zero;

    for (int c = 0; c < IN_DIM / 4; ++c) {
        const int kk = c * 4;
        v2f a[4], b[4];
#pragma unroll
        for (int ms = 0; ms < 4; ++ms)
            a[ms] = *(const v2f*)(inputs + aBase[ms] + kk);
#pragma unroll
        for (int nt = 0; nt < 4; ++nt)
            b[nt] = *(const v2f*)(Wp2f + (size_t)(bBase[nt] + c * 1024) * 2);
#pragma unroll
        for (int ms = 0; ms < 4; ++ms)
#pragma unroll
            for (int nt = 0; nt < 4; ++nt)
                acc[ms][nt] = __builtin_amdgcn_wmma_f32_16x16x4_f32(
                    false, a[ms], false, b[nt], (short)0, acc[ms][nt],
                    false, false);
    }

    // -------- fused LRU epilogue (wave-uniform re/im split) --------
    const int isIm = (n0 >= HID);
    const int h0   = n0 & 255;

    float lamR[4], lamI[4];
#pragma unroll
    for (int nt = 0; nt < 4; ++nt) {
        int h = h0 + nt * 16 + llo;
        lamR[nt] = lam[h];
        lamI[nt] = lam[HID + h];
    }

    float* outp = out + (size_t)isIm * (size_t)B_SZ * HID;

#pragma unroll
    for (int ms = 0; ms < 4; ++ms) {
#pragma unroll
        for (int r = 0; r < 8; ++r) {
            const int row = m0 + ms * 16 + lhi * 8 + r;   // C layout: lanes>=16 hold M=8+r
#pragma unroll
            for (int nt = 0; nt < 4; ++nt) {
                const int h = h0 + nt * 16 + llo;
                const size_t idx = (size_t)row * HID + h;
                const float hr = h_re[idx];
                const float hi = h_im[idx];
                const float proj = acc[ms][nt][r];
                const float v = isIm ? fmaf(lamR[nt], hi, fmaf(lamI[nt], hr, proj))
                                     : fmaf(lamR[nt], hr, fmaf(-lamI[nt], hi, proj));
                outp[idx] = v;
            }
        }
    }
}

// ---------------------------------------------------------------------------
extern "C" void kernel_launch(void* const* d_in, const int* in_sizes, int n_in,
                              void* d_out, int out_size, void* d_ws, size_t ws_size,
                              hipStream_t stream) {
    const float* inputs    = (const float*)d_in[0];
    const float* h_re      = (const float*)d_in[1];
    const float* h_im      = (const float*)d_in[2];
    const float* nu_log    = (const float*)d_in[3];
    const float* theta_log = (const float*)d_in[4];
    const float* B_real    = (const float*)d_in[5];
    const float* B_img     = (const float*)d_in[6];
    const float* gamma_log = (const float*)d_in[7];

    float2* Wp2 = (float2*)d_ws;                 // 32768 * 8B = 256 KB
    float*  lam = (float*)d_ws + 32768 * 2;      // 512 floats

    lru_prep<<<(32768 + HID + 255) / 256, 256, 0, stream>>>(
        nu_log, theta_log, B_real, B_img, gamma_log, Wp2, lam);

    lru_main<<<B_SZ / 64, 256, 0, stream>>>(
        inputs, h_re, h_im, (const float*)d_ws, lam, (float*)d_out);
}